// GraphTransformerEncoderLayer_65687229825300
// MI455X (gfx1250) — compile-verified
//
#include <hip/hip_runtime.h>
#include <hip/hip_bf16.h>

// ---------------- problem constants ----------------
#define NN 50000          // nodes
#define NE 800000         // edges
#define GG 64             // graphs
#define SLD 132           // padded LDS row stride (floats) -> conflict-free A-frag reads

typedef __attribute__((ext_vector_type(2))) float v2f;
typedef __attribute__((ext_vector_type(8))) float v8f;

// D = A(16x4) * B(4x16) + C  — native f32 WMMA (full precision path)
__device__ inline v8f wmma4(v2f a, v2f b, v8f c) {
  return __builtin_amdgcn_wmma_f32_16x16x4_f32(false, a, false, b, (short)0, c,
                                               false, false);
}

// one k-step (K=4) against 8 adjacent 16-col output tiles.
// bp must already point at  B + (2*laneHalf)*ldb + colBase + laneN
__device__ inline void wmma_row8(v2f a, const float* __restrict__ bp, int ldb,
                                 int ks, v8f c[8]) {
  const float* bk = bp + (size_t)(4 * ks) * ldb;
#pragma unroll
  for (int t = 0; t < 8; ++t) {
    v2f b;
    b.x = bk[t * 16];
    b.y = bk[t * 16 + ldb];
    c[t] = wmma4(a, b, c[t]);
  }
}

// same, but each loaded B fragment feeds TWO A tiles (doubles wmma per B load)
__device__ inline void wmma_row8x2(v2f a0, v2f a1, const float* __restrict__ bp,
                                   int ldb, int ks, v8f c0[8], v8f c1[8]) {
  const float* bk = bp + (size_t)(4 * ks) * ldb;
#pragma unroll
  for (int t = 0; t < 8; ++t) {
    v2f b;
    b.x = bk[t * 16];
    b.y = bk[t * 16 + ldb];
    c0[t] = wmma4(a0, b, c0[t]);
    c1[t] = wmma4(a1, b, c1[t]);
  }
}

// wave-local LDS ordering: hardware keeps DS ops in order per wave; the wait +
// memory clobber stops compiler reordering and ensures returned data is ready.
__device__ inline void lds_fence() {
  asm volatile("s_wait_dscnt 0" ::: "memory");
}

// ================= kernel 1: Qh / Kh / Vh = x @ {Wq,Wk,Wv} =================
__global__ __launch_bounds__(256) void qkv_kernel(
    const float* __restrict__ x, const float* __restrict__ Wq,
    const float* __restrict__ bq, const float* __restrict__ Wk,
    const float* __restrict__ Wv, float* __restrict__ Qh,
    float* __restrict__ Kh, float* __restrict__ Vh) {
  int wave = (blockIdx.x * blockDim.x + threadIdx.x) >> 5;
  int lane = threadIdx.x & 31;
  int lh = lane >> 4, ln = lane & 15;
  if (wave >= NN / 16) return;
  int rb = wave * 16;
  const float* ap0 = x + (size_t)(rb + ln) * 128 + 2 * lh;
#pragma unroll
  for (int mm = 0; mm < 3; ++mm) {
    const float* W = (mm == 0) ? Wq : (mm == 1) ? Wk : Wv;
    float* Out = (mm == 0) ? Qh : (mm == 1) ? Kh : Vh;
    v8f c[8] = {};
    const float* bp = W + 2 * lh * 128 + ln;
#pragma unroll 4
    for (int ks = 0; ks < 32; ++ks) {
      v2f a = *(const v2f*)(ap0 + 4 * ks);
      wmma_row8(a, bp, 128, ks, c);
    }
#pragma unroll
    for (int t = 0; t < 8; ++t) {
      int col = t * 16 + ln;
      float bias = (mm == 0) ? bq[col] : 0.f;
#pragma unroll
      for (int v = 0; v < 8; ++v)
        Out[(size_t)(rb + v + 8 * lh) * 128 + col] = c[t][v] + bias;
    }
  }
}

// ================= kernel 2: th/te = relu(temb) @ Wt_{h,e} + bt =============
__global__ __launch_bounds__(256) void temb_kernel(
    const float* __restrict__ temb, const float* __restrict__ Wt_h,
    const float* __restrict__ bt_h, const float* __restrict__ Wt_e,
    const float* __restrict__ bt_e, float* __restrict__ th,
    float* __restrict__ te) {
  int w = threadIdx.x >> 5;  // 8 waves: 4 row-tiles x 2 matrices
  int lane = threadIdx.x & 31;
  int lh = lane >> 4, ln = lane & 15;
  int rb = (w & 3) * 16;
  const float* W = (w < 4) ? Wt_h : Wt_e;
  const float* bias = (w < 4) ? bt_h : bt_e;
  float* Out = (w < 4) ? th : te;
  v8f c[8] = {};
  const float* ap0 = temb + (size_t)(rb + ln) * 128 + 2 * lh;
  const float* bp = W + 2 * lh * 128 + ln;
#pragma unroll 4
  for (int ks = 0; ks < 32; ++ks) {
    v2f a = *(const v2f*)(ap0 + 4 * ks);
    a.x = fmaxf(a.x, 0.f);
    a.y = fmaxf(a.y, 0.f);
    wmma_row8(a, bp, 128, ks, c);
  }
#pragma unroll
  for (int t = 0; t < 8; ++t) {
    int col = t * 16 + ln;
#pragma unroll
    for (int v = 0; v < 8; ++v)
      Out[(size_t)(rb + v + 8 * lh) * 128 + col] = c[t][v] + bias[col];
  }
}

// ================= kernel 3: fused edge pass (32 edges / wave) ==============
// Two 16-edge subtiles per wave share every B fragment: halves weight traffic.
// Ee GEMM -> score (LDS) -> exp/atomics -> e GEMM -> +bias +te[batch[src]]
// +edge_attr residual -> LayerNorm -> e_out.
__global__ __launch_bounds__(96) void edge_kernel(
    const float* __restrict__ edge_attr, const int* __restrict__ eidx,
    const int* __restrict__ batch_vec, const float* __restrict__ We,
    const float* __restrict__ be, const float* __restrict__ Wo_e,
    const float* __restrict__ bo_e, const float* __restrict__ g1e,
    const float* __restrict__ b1e, const float* __restrict__ Qh,
    const float* __restrict__ Kh, const float* __restrict__ Vh,
    const float* __restrict__ te, float* __restrict__ ssum,
    float* __restrict__ accv, float* __restrict__ e_out) {
  __shared__ float S[3][2][16 * SLD];  // 50688 B
  __shared__ int IS[3][32], ID[3][32], IB[3][32];
  __shared__ float EX[3][256];
  __shared__ float MU[3][2][32];

  int w = threadIdx.x >> 5;
  int lane = threadIdx.x & 31;
  int lh = lane >> 4, ln = lane & 15;
  int tile = blockIdx.x * 3 + w;
  if (tile >= NE / 32) return;
  int tb = tile * 32;

  {  // stage 32 src/dst/batch indices
    int s = eidx[tb + lane];
    IS[w][lane] = s;
    ID[w][lane] = eidx[NE + tb + lane];
    IB[w][lane] = batch_vec[s];
  }
  lds_fence();

  const float* ap0 = edge_attr + (size_t)(tb + ln) * 128 + 2 * lh;
  const float* ap1 = edge_attr + (size_t)(tb + 16 + ln) * 128 + 2 * lh;
  __builtin_prefetch(ap0, 0, 1);
  __builtin_prefetch(ap1, 0, 1);

  // ---- Ee = edge_attr @ We (+be) fused into score = K*Q*Ew + Eb ----
#pragma unroll
  for (int p = 0; p < 2; ++p) {  // heads 0-3 then 4-7 (w,b tile pairs)
    v8f c0[8] = {}, c1[8] = {};
    const float* bp = We + 2 * lh * 256 + p * 128 + ln;
#pragma unroll 4
    for (int ks = 0; ks < 32; ++ks) {
      v2f a0 = *(const v2f*)(ap0 + 4 * ks);
      v2f a1 = *(const v2f*)(ap1 + 4 * ks);
      wmma_row8x2(a0, a1, bp, 256, ks, c0, c1);
    }
#pragma unroll
    for (int q = 0; q < 4; ++q) {
      int h = p * 4 + q;
      float bw = be[p * 128 + q * 32 + ln];
      float bb = be[p * 128 + q * 32 + 16 + ln];
#pragma unroll
      for (int v = 0; v < 8; ++v) {
        int m = v + 8 * lh;
        {  // subtile 0
          float Kv = Kh[(size_t)IS[w][m] * 128 + h * 16 + ln];
          float Qv = Qh[(size_t)ID[w][m] * 128 + h * 16 + ln];
          S[w][0][m * SLD + h * 16 + ln] =
              Kv * Qv * (c0[2 * q][v] + bw) + (c0[2 * q + 1][v] + bb);
        }
        {  // subtile 1
          float Kv = Kh[(size_t)IS[w][16 + m] * 128 + h * 16 + ln];
          float Qv = Qh[(size_t)ID[w][16 + m] * 128 + h * 16 + ln];
          S[w][1][m * SLD + h * 16 + ln] =
              Kv * Qv * (c1[2 * q][v] + bw) + (c1[2 * q + 1][v] + bb);
        }
      }
    }
  }
  lds_fence();

  // ---- ex = exp(clip(sum_d(score)/4)); s in [-5,5] => max-pass not needed --
#pragma unroll
  for (int r = 0; r < 8; ++r) {
    int idx = r * 32 + lane;          // 256 (edge,head) pairs
    int m32 = idx >> 3, h = idx & 7;
    const float* row = &S[w][m32 >> 4][(m32 & 15) * SLD + h * 16];
    float s = 0.f;
#pragma unroll
    for (int d = 0; d < 16; ++d) s += row[d];
    s = fminf(5.f, fmaxf(-5.f, s * 0.25f));
    float ex = __expf(s);
    EX[w][idx] = ex;
    atomicAdd(&ssum[(size_t)ID[w][m32] * 8 + h], ex);
  }
  lds_fence();

  // ---- softmax numerator scatter: acc[dst] += (Vh[src]+score)*ex ----
  for (int m32 = 0; m32 < 32; ++m32) {
    int sN = IS[w][m32], dN = ID[w][m32];
    const float* row = &S[w][m32 >> 4][(m32 & 15) * SLD];
#pragma unroll
    for (int r = 0; r < 4; ++r) {
      int j = r * 32 + lane;
      float ex = EX[w][m32 * 8 + (j >> 4)];
      float val = (Vh[(size_t)sN * 128 + j] + row[j]) * ex;
      atomicAdd(&accv[(size_t)dN * 128 + j], val);
    }
  }

  // ---- e = score @ Wo_e (A-fragments from LDS, B shared across subtiles) ---
  v8f c0[8] = {}, c1[8] = {};
  {
    const float* bp = Wo_e + 2 * lh * 128 + ln;
    const float* sp0 = &S[w][0][ln * SLD + 2 * lh];
    const float* sp1 = &S[w][1][ln * SLD + 2 * lh];
#pragma unroll 4
    for (int ks = 0; ks < 32; ++ks) {
      v2f a0, a1;
      a0.x = sp0[4 * ks]; a0.y = sp0[4 * ks + 1];
      a1.x = sp1[4 * ks]; a1.y = sp1[4 * ks + 1];
      wmma_row8x2(a0, a1, bp, 128, ks, c0, c1);
    }
  }
  lds_fence();
  // ---- residual + biases into LDS (pre-LayerNorm) ----
#pragma unroll
  for (int t = 0; t < 8; ++t) {
    int col = t * 16 + ln;
    float bo = bo_e[col];
#pragma unroll
    for (int v = 0; v < 8; ++v) {
      int m = v + 8 * lh;
      S[w][0][m * SLD + col] = c0[t][v] + bo +
                               te[(size_t)IB[w][m] * 128 + col] +
                               edge_attr[(size_t)(tb + m) * 128 + col];
      S[w][1][m * SLD + col] = c1[t][v] + bo +
                               te[(size_t)IB[w][16 + m] * 128 + col] +
                               edge_attr[(size_t)(tb + 16 + m) * 128 + col];
    }
  }
  lds_fence();
  // ---- LayerNorm (per-row mean/var: 2 lanes per row) + coalesced store ----
#pragma unroll
  for (int u = 0; u < 2; ++u) {
    {
      int r = lane >> 1, off = (lane & 1) * 64;
      float s = 0.f, q = 0.f;
      for (int j = 0; j < 64; ++j) {
        float v = S[w][u][r * SLD + off + j];
        s += v;
        q += v * v;
      }
      s += __shfl_xor(s, 1, 32);
      q += __shfl_xor(q, 1, 32);
      float mu = s * (1.f / 128.f);
      float var = q * (1.f / 128.f) - mu * mu;
      if ((lane & 1) == 0) {
        MU[w][u][r] = mu;
        MU[w][u][16 + r] = rsqrtf(var + 1e-5f);
      }
    }
    lds_fence();
    for (int it = 0; it < 64; ++it) {  // coalesced 128B stores
      int idx = it * 32 + lane;
      int m = idx >> 7, col = idx & 127;
      float vv = (S[w][u][m * SLD + col] - MU[w][u][m]) * MU[w][u][16 + m];
      e_out[(size_t)(tb + u * 16) * 128 + idx] = vv * g1e[col] + b1e[col];
    }
  }
}

// ================= kernel 4: node finalize ==================================
// wV = acc/ssum ; h = LN2( LN1(x + wV@Wo_h + bo_h + th[b]) + FFN )
__global__ __launch_bounds__(64) void node_kernel(
    const float* __restrict__ x, const int* __restrict__ batch_vec,
    const float* __restrict__ Wo_h, const float* __restrict__ bo_h,
    const float* __restrict__ th, const float* __restrict__ g1h,
    const float* __restrict__ b1h, const float* __restrict__ Wf1h,
    const float* __restrict__ bf1h, const float* __restrict__ Wf2h,
    const float* __restrict__ bf2h, const float* __restrict__ g2h,
    const float* __restrict__ b2h, const float* __restrict__ ssum,
    const float* __restrict__ accv, float* __restrict__ h_out) {
  __shared__ float S[2][16 * SLD];
  __shared__ float F[2][16 * SLD];
  __shared__ int IB[2][16];
  __shared__ float MU[2][32];

  int w = threadIdx.x >> 5;
  int lane = threadIdx.x & 31;
  int lh = lane >> 4, ln = lane & 15;
  int tile = blockIdx.x * 2 + w;
  if (tile >= NN / 16) return;
  int rb = tile * 16;
  float* Sw = S[w];
  float* Fw = F[w];

  if (lane < 16) IB[w][lane] = batch_vec[rb + lane];
  lds_fence();

  // ---- (acc * 1/(ssum+eps)) @ Wo_h ----
  v8f c[8] = {};
  {
    const float* ap0 = accv + (size_t)(rb + ln) * 128 + 2 * lh;
    const float* sr = ssum + (size_t)(rb + ln) * 8;
    const float* bp = Wo_h + 2 * lh * 128 + ln;
#pragma unroll 4
    for (int ks = 0; ks < 32; ++ks) {
      int kb = 4 * ks + 2 * lh;
      float inv = 1.f / (sr[kb >> 4] + 1e-16f);
      v2f a = *(const v2f*)(ap0 + 4 * ks);
      a.x *= inv;
      a.y *= inv;
      wmma_row8(a, bp, 128, ks, c);
    }
  }
#pragma unroll
  for (int t = 0; t < 8; ++t) {
    int col = t * 16 + ln;
    float bo = bo_h[col];
#pragma unroll
    for (int v = 0; v < 8; ++v) {
      int m = v + 8 * lh;
      Sw[m * SLD + col] = c[t][v] + bo + th[(size_t)IB[w][m] * 128 + col] +
                          x[(size_t)(rb + m) * 128 + col];
    }
  }
  lds_fence();
  // ---- LN1 in place ----
  {
    int r = lane >> 1, off = (lane & 1) * 64;
    float s = 0.f, q = 0.f;
    for (int j = 0; j < 64; ++j) {
      float v = Sw[r * SLD + off + j];
      s += v;
      q += v * v;
    }
    s += __shfl_xor(s, 1, 32);
    q += __shfl_xor(q, 1, 32);
    float mu = s * (1.f / 128.f);
    float var = q * (1.f / 128.f) - mu * mu;
    if ((lane & 1) == 0) {
      MU[w][r] = mu;
      MU[w][16 + r] = rsqrtf(var + 1e-5f);
    }
  }
  lds_fence();
  for (int it = 0; it < 64; ++it) {
    int idx = it * 32 + lane;
    int m = idx >> 7, col = idx & 127;
    Sw[m * SLD + col] =
        (Sw[m * SLD + col] - MU[w][m]) * MU[w][16 + m] * g1h[col] + b1h[col];
  }
  lds_fence();

  // ---- FFN: h2 = relu(h@Wf1h+bf1h)@Wf2h, k split in two 128 halves ----
  v8f h2[8] = {};
#pragma unroll
  for (int p = 0; p < 2; ++p) {
    v8f f[8] = {};
    {
      const float* bp = Wf1h + 2 * lh * 256 + p * 128 + ln;
      const float* sp = &Sw[ln * SLD + 2 * lh];
#pragma unroll 4
      for (int ks = 0; ks < 32; ++ks) {
        v2f a;
        a.x = sp[4 * ks];
        a.y = sp[4 * ks + 1];
        wmma_row8(a, bp, 256, ks, f);
      }
    }
    lds_fence();  // prior-iteration Fw reads done before overwrite
#pragma unroll
    for (int t = 0; t < 8; ++t) {
      int gcol = p * 128 + t * 16 + ln;
      float bf = bf1h[gcol];
#pragma unroll
      for (int v = 0; v < 8; ++v) {
        int m = v + 8 * lh;
        Fw[m * SLD + t * 16 + ln] = fmaxf(f[t][v] + bf, 0.f);
      }
    }
    lds_fence();
    {
      const float* bp = Wf2h + (size_t)(p * 128 + 2 * lh) * 128 + ln;
      const float* fp = &Fw[ln * SLD + 2 * lh];
#pragma unroll 4
      for (int ks = 0; ks < 32; ++ks) {
        v2f a;
        a.x = fp[4 * ks];
        a.y = fp[4 * ks + 1];
        wmma_row8(a, bp, 128, ks, h2);
      }
    }
  }
  lds_fence();
  // ---- residual + LN2 -> h_out ----
#pragma unroll
  for (int t = 0; t < 8; ++t) {
    int col = t * 16 + ln;
    float bf = bf2h[col];
#pragma unroll
    for (int v = 0; v < 8; ++v) {
      int m = v + 8 * lh;
      Fw[m * SLD + col] = h2[t][v] + bf + Sw[m * SLD + col];
    }
  }
  lds_fence();
  {
    int r = lane >> 1, off = (lane & 1) * 64;
    float s = 0.f, q = 0.f;
    for (int j = 0; j < 64; ++j) {
      float v = Fw[r * SLD + off + j];
      s += v;
      q += v * v;
    }
    s += __shfl_xor(s, 1, 32);
    q += __shfl_xor(q, 1, 32);
    float mu = s * (1.f / 128.f);
    float var = q * (1.f / 128.f) - mu * mu;
    if ((lane & 1) == 0) {
      MU[w][r] = mu;
      MU[w][16 + r] = rsqrtf(var + 1e-5f);
    }
  }
  lds_fence();
  for (int it = 0; it < 64; ++it) {
    int idx = it * 32 + lane;
    int m = idx >> 7, col = idx & 127;
    float vv = (Fw[m * SLD + col] - MU[w][m]) * MU[w][16 + m];
    h_out[(size_t)rb * 128 + idx] = vv * g2h[col] + b2h[col];
  }
}

// ================= host launch ==============================================
extern "C" void kernel_launch(void* const* d_in, const int* in_sizes, int n_in,
                              void* d_out, int out_size, void* d_ws,
                              size_t ws_size, hipStream_t stream) {
  const float* x = (const float*)d_in[0];
  const float* edge_attr = (const float*)d_in[1];
  const float* temb = (const float*)d_in[2];
  const int* eidx = (const int*)d_in[3];
  const int* batch_vec = (const int*)d_in[4];
  const float* Wq = (const float*)d_in[5];
  const float* bq = (const float*)d_in[6];
  const float* Wk = (const float*)d_in[7];
  const float* Wv = (const float*)d_in[8];
  const float* We = (const float*)d_in[9];
  const float* be = (const float*)d_in[10];
  const float* Wo_h = (const float*)d_in[11];
  const float* bo_h = (const float*)d_in[12];
  const float* Wo_e = (const float*)d_in[13];
  const float* bo_e = (const float*)d_in[14];
  const float* Wt_h = (const float*)d_in[15];
  const float* bt_h = (const float*)d_in[16];
  const float* Wt_e = (const float*)d_in[17];
  const float* bt_e = (const float*)d_in[18];
  const float* g1h = (const float*)d_in[19];
  const float* b1h = (const float*)d_in[20];
  const float* g1e = (const float*)d_in[21];
  const float* b1e = (const float*)d_in[22];
  const float* Wf1h = (const float*)d_in[23];
  const float* bf1h = (const float*)d_in[24];
  const float* Wf2h = (const float*)d_in[25];
  const float* bf2h = (const float*)d_in[26];
  const float* g2h = (const float*)d_in[27];
  const float* b2h = (const float*)d_in[28];

  float* ws = (float*)d_ws;
  float* Qh = ws;
  float* Kh = Qh + (size_t)NN * 128;
  float* Vh = Kh + (size_t)NN * 128;
  float* accv = Vh + (size_t)NN * 128;
  float* ssum = accv + (size_t)NN * 128;  // N*8
  float* th = ssum + (size_t)NN * 8;      // G*128
  float* te = th + (size_t)GG * 128;

  float* h_out = (float*)d_out;
  float* e_out = h_out + (size_t)NN * 128;

  // zero the scatter accumulators every call (graph-capture safe memset node)
  hipMemsetAsync(accv, 0, ((size_t)NN * 128 + (size_t)NN * 8) * sizeof(float),
                 stream);

  qkv_kernel<<<(NN / 16 + 7) / 8, 256, 0, stream>>>(x, Wq, bq, Wk, Wv, Qh, Kh,
                                                    Vh);
  temb_kernel<<<1, 256, 0, stream>>>(temb, Wt_h, bt_h, Wt_e, bt_e, th, te);
  edge_kernel<<<(NE / 32 + 2) / 3, 96, 0, stream>>>(
      edge_attr, eidx, batch_vec, We, be, Wo_e, bo_e, g1e, b1e, Qh, Kh, Vh, te,
      ssum, accv, e_out);
  node_kernel<<<(NN / 16 + 1) / 2, 64, 0, stream>>>(
      x, batch_vec, Wo_h, bo_h, th, g1h, b1h, Wf1h, bf1h, Wf2h, bf2h, g2h, b2h,
      ssum, accv, h_out);
}